// TokenMergeModule_79070347919760
// MI455X (gfx1250) — compile-verified
//
#include <hip/hip_runtime.h>
#include <hip/hip_bf16.h>

// ---------------- problem constants (fixed shapes from reference) ----------
#define Bn    4
#define Sn    4096
#define Dn    512
#define Gn    64
#define Nn    4096
#define Rn    1024
#define WIN   16
#define KEEP  (Sn - Rn)          // 3072
#define NPAIR 61320              // sum_{d=1..15} (S-d)
#define PPAD  65536              // pow2 pad for bitonic sort
#define TB    4096               // bitonic LDS tile (u64 keys) = 32KB
#define WLDSTRIDE 68             // 64 + 4 pad: b128-aligned, conflict-free

typedef __attribute__((ext_vector_type(2))) float v2f;
typedef __attribute__((ext_vector_type(4))) float f4;
typedef __attribute__((ext_vector_type(8))) float v8f;

// ========================================================================
// Kernel 1: g = x @ W via V_WMMA_F32_16X16X4_F32, fused row-normalize.
// W (512x64 f32) is staged once per block into dynamic LDS (row stride 68
// floats: keeps ds_store_b128 16B-aligned and the two 16-lane halves on
// disjoint banks). One wave computes a 16(M) x 64(N) tile; K=512 step 4.
// A frag (16x4, 2 VGPR): lane L -> row m0+(L%16), K = k + 2*(L/16)+v.
// B frag (4x16, 2 VGPR): lane L -> col n0+(L%16), K = k + v + 2*(L/16).
// C/D (16x16, 8 VGPR):   lane L, reg c -> row m0+c+8*(L/16), col n0+(L%16).
// ========================================================================
__global__ void gemm_norm_kernel(const float* __restrict__ x,
                                 const float* __restrict__ W,
                                 float* __restrict__ gn,
                                 float* __restrict__ norms) {
  extern __shared__ float wlds[];  // Dn * WLDSTRIDE floats = 139264 B

  // cooperative stage of W: 512 rows x 16 float4 = 8192 float4 / 256 thr
  for (int f = threadIdx.x; f < Dn * (Gn / 4); f += blockDim.x) {
    int row = f >> 4;            // 16 float4 per row
    int c4  = (f & 15) << 2;
    *(f4*)&wlds[row * WLDSTRIDE + c4] = *(const f4*)(W + row * Gn + c4);
  }
  __syncthreads();

  const int lane = threadIdx.x & 31;
  const int wave = threadIdx.x >> 5;
  const int m0   = (blockIdx.x * 8 + wave) * 16;
  const int l16  = lane & 15;
  const int h    = lane >> 4;

  const float* xRow  = x + (size_t)(m0 + l16) * Dn + 2 * h;
  const float* wBase = wlds + l16;

  v8f acc[4] = {};  // four 16x16 N-tiles covering all 64 columns

  for (int k = 0; k < Dn; k += 4) {
    v2f a = *(const v2f*)(xRow + k);                       // 8B pair load
    const float* wr = wBase + (size_t)(k + 2 * h) * WLDSTRIDE;
#pragma unroll
    for (int t = 0; t < 4; ++t) {
      v2f b;
      b.x = wr[16 * t];               // K = k + 2h
      b.y = wr[WLDSTRIDE + 16 * t];   // K = k + 2h + 1
      acc[t] = __builtin_amdgcn_wmma_f32_16x16x4_f32(
          false, a, false, b, (short)0, acc[t], false, false);
    }
  }

  // Row norms + normalize. Row (m0 + c + 8h) lives in reg index c across
  // lanes of one 16-lane half; reduce with wave32 xor shuffles 1,2,4,8.
#pragma unroll
  for (int c = 0; c < 8; ++c) {
    float ss = 0.0f;
#pragma unroll
    for (int t = 0; t < 4; ++t) { float v = acc[t][c]; ss += v * v; }
    ss += __shfl_xor(ss, 1, 32);
    ss += __shfl_xor(ss, 2, 32);
    ss += __shfl_xor(ss, 4, 32);
    ss += __shfl_xor(ss, 8, 32);
    float nrm = sqrtf(ss);
    float inv = 1.0f / fmaxf(nrm, 1e-12f);
    int row = m0 + c + 8 * h;
#pragma unroll
    for (int t = 0; t < 4; ++t)
      gn[(size_t)row * Gn + 16 * t + l16] = acc[t][c] * inv;
    if (l16 == 0) norms[row] = nrm;
  }
}

// ========================================================================
// Kernel 2: cosine sim per candidate pair -> 64-bit sort key.
// key = (~orderedBits(sim) << 32) | (i*15 + (d-1)); ascending sort gives
// score DESC, then i ASC, then j ASC (matches lexsort((pj, pi, -sim))).
// ========================================================================
__global__ void sim_kernel(const float* __restrict__ gn,
                           unsigned long long* __restrict__ keys) {
  int t = blockIdx.x * blockDim.x + threadIdx.x;
  int b = t >> 16;
  int p = t & (PPAD - 1);
  unsigned long long key = ~0ull;  // pad: sorts last
  if (p < NPAIR) {
    int rem = p, d = 1;
    while (rem >= Sn - d) { rem -= Sn - d; ++d; }   // <=15 iters
    int i = rem, j = i + d;
    const f4* gi = (const f4*)(gn + ((size_t)b * Sn + i) * Gn);
    const f4* gj = (const f4*)(gn + ((size_t)b * Sn + j) * Gn);
    float acc = 0.0f;
#pragma unroll
    for (int c = 0; c < Gn / 4; ++c) {
      f4 a = gi[c], bb = gj[c];
      acc += a.x * bb.x + a.y * bb.y + a.z * bb.z + a.w * bb.w;
    }
    unsigned int u = __float_as_uint(acc);
    u ^= (u >> 31) ? 0xFFFFFFFFu : 0x80000000u;     // monotone map (ASC)
    unsigned int hi = ~u;                           // DESC
    unsigned int pid = (unsigned int)(i * (WIN - 1) + (d - 1));
    key = ((unsigned long long)hi << 32) | pid;
  }
  keys[((size_t)b << 16) + p] = key;
}

// ========================================================================
// Kernel 3a: LDS-fused local bitonic sort of TB-element tiles: all stages
// k = 2..TB done in one launch (tiles never straddle a batch: PPAD % TB == 0).
// ========================================================================
__global__ void bitonic_local_sort_kernel(unsigned long long* __restrict__ keys) {
  __shared__ unsigned long long tile[TB];
  size_t base = (size_t)blockIdx.x * TB;
  int ib0 = (int)(base & (PPAD - 1));               // index within batch
  for (int t = threadIdx.x; t < TB; t += blockDim.x) tile[t] = keys[base + t];
  __syncthreads();
  for (int k = 2; k <= TB; k <<= 1) {
    for (int j = k >> 1; j > 0; j >>= 1) {
      for (int p = threadIdx.x; p < TB / 2; p += blockDim.x) {
        int i   = ((p & ~(j - 1)) << 1) | (p & (j - 1));
        int ixj = i | j;
        unsigned long long a = tile[i], c = tile[ixj];
        bool up = (((ib0 + i) & k) == 0);
        if ((a > c) == up) { tile[i] = c; tile[ixj] = a; }
      }
      __syncthreads();
    }
  }
  for (int t = threadIdx.x; t < TB; t += blockDim.x) keys[base + t] = tile[t];
}

// Kernel 3b: single global bitonic step (strides j >= TB).
__global__ void bitonic_global_kernel(unsigned long long* __restrict__ keys,
                                      int k, int j) {
  int t = blockIdx.x * blockDim.x + threadIdx.x;
  int b = t >> 16;
  int i = t & (PPAD - 1);
  int ixj = i ^ j;
  if (ixj > i) {
    unsigned long long* kb = keys + ((size_t)b << 16);
    unsigned long long a = kb[i], c = kb[ixj];
    bool up = ((i & k) == 0);
    if ((a > c) == up) { kb[i] = c; kb[ixj] = a; }
  }
}

// Kernel 3c: LDS-fused merge: all strides j = TB/2..1 for a given outer k.
__global__ void bitonic_local_merge_kernel(unsigned long long* __restrict__ keys,
                                           int k) {
  __shared__ unsigned long long tile[TB];
  size_t base = (size_t)blockIdx.x * TB;
  int ib0 = (int)(base & (PPAD - 1));
  for (int t = threadIdx.x; t < TB; t += blockDim.x) tile[t] = keys[base + t];
  __syncthreads();
  for (int j = TB >> 1; j > 0; j >>= 1) {
    for (int p = threadIdx.x; p < TB / 2; p += blockDim.x) {
      int i   = ((p & ~(j - 1)) << 1) | (p & (j - 1));
      int ixj = i | j;
      unsigned long long a = tile[i], c = tile[ixj];
      bool up = (((ib0 + i) & k) == 0);
      if ((a > c) == up) { tile[i] = c; tile[ixj] = a; }
    }
    __syncthreads();
  }
  for (int t = threadIdx.x; t < TB; t += blockDim.x) keys[base + t] = tile[t];
}

// ========================================================================
// Kernel 4: serial greedy matching (inherently sequential) + keep map.
// One block per batch; lane 0 scans sorted keys with LDS used[]/skip[].
// ========================================================================
__global__ void greedy_kernel(const unsigned long long* __restrict__ keys,
                              int* __restrict__ partner,
                              int* __restrict__ keepmap) {
  int b = blockIdx.x;
  __shared__ unsigned char used[Sn];
  __shared__ unsigned char skip[Sn];
  for (int s = threadIdx.x; s < Sn; s += blockDim.x) {
    used[s] = 0; skip[s] = 0; partner[b * Sn + s] = -1;
  }
  __syncthreads();
  if (threadIdx.x == 0) {
    const unsigned long long* kb = keys + ((size_t)b << 16);
    int cnt = 0;
    for (int t = 0; t < PPAD && cnt < Rn; ++t) {
      unsigned long long key = kb[t];
      unsigned int lo = (unsigned int)key;
      if (lo == 0xFFFFFFFFu) continue;              // padding
      int pid = (int)lo;
      int i = pid / (WIN - 1);
      int d = pid % (WIN - 1) + 1;
      int j = i + d;
      if (j >= Sn) continue;
      if (used[i] || used[j]) continue;
      used[i] = 1; used[j] = 1; skip[j] = 1;
      partner[b * Sn + i] = j;
      ++cnt;
    }
    // stable keep: first S-r rows with skip==0 (matches argsort truncation)
    int pos = 0;
    for (int s = 0; s < Sn && pos < KEEP; ++s)
      if (!skip[s]) keepmap[b * KEEP + pos++] = s;
  }
}

// ========================================================================
// Kernel 5: merge x rows (norm-weighted) + gather kept position ids.
// ========================================================================
__global__ void merge_x_kernel(const float* __restrict__ x,
                               const float* __restrict__ norms,
                               const int* __restrict__ partner,
                               const int* __restrict__ keepmap,
                               const int* __restrict__ pos_in,
                               float* __restrict__ out_x,
                               int* __restrict__ out_p) {
  int kr = blockIdx.x % KEEP;
  int b  = blockIdx.x / KEEP;
  int row = keepmap[b * KEEP + kr];
  int p   = partner[b * Sn + row];
  if (threadIdx.x == 0) out_p[b * KEEP + kr] = pos_in[b * Sn + row];

  const f4* a4 = (const f4*)(x + ((size_t)b * Sn + row) * Dn);
  f4* o4 = (f4*)(out_x + ((size_t)b * KEEP + kr) * Dn);
  if (p >= 0) {
    float wi = norms[b * Sn + row];
    float wj = norms[b * Sn + p];
    float inv = 1.0f / (wi + wj + 1e-8f);
    const f4* b4 = (const f4*)(x + ((size_t)b * Sn + p) * Dn);
    for (int c = threadIdx.x; c < Dn / 4; c += blockDim.x) {
      f4 av = a4[c], bv = b4[c];
      f4 r;
      r.x = (wi * av.x + wj * bv.x) * inv;
      r.y = (wi * av.y + wj * bv.y) * inv;
      r.z = (wi * av.z + wj * bv.z) * inv;
      r.w = (wi * av.w + wj * bv.w) * inv;
      o4[c] = r;
    }
  } else {
    for (int c = threadIdx.x; c < Dn / 4; c += blockDim.x) o4[c] = a4[c];
  }
}

// ========================================================================
// Kernel 6: merge source rows — the bandwidth-dominant streaming pass
// (~0.45 GB). Non-temporal loads/stores (TH_NT) to stream past caches.
// ========================================================================
__global__ void merge_s_kernel(const float* __restrict__ src,
                               const int* __restrict__ partner,
                               const int* __restrict__ keepmap,
                               float* __restrict__ out_s) {
  int kr = blockIdx.x % KEEP;
  int b  = blockIdx.x / KEEP;
  int row = keepmap[b * KEEP + kr];
  int p   = partner[b * Sn + row];
  const f4* a4 = (const f4*)(src + ((size_t)b * Sn + row) * Nn);
  f4* o4 = (f4*)(out_s + ((size_t)b * KEEP + kr) * Nn);
  if (p >= 0) {
    const f4* b4 = (const f4*)(src + ((size_t)b * Sn + p) * Nn);
    for (int c = threadIdx.x; c < Nn / 4; c += blockDim.x) {
      f4 av = __builtin_nontemporal_load(&a4[c]);
      f4 bv = __builtin_nontemporal_load(&b4[c]);
      f4 r = av + bv;
      __builtin_nontemporal_store(r, &o4[c]);
    }
  } else {
    for (int c = threadIdx.x; c < Nn / 4; c += blockDim.x) {
      f4 av = __builtin_nontemporal_load(&a4[c]);
      __builtin_nontemporal_store(av, &o4[c]);
    }
  }
}

// ========================================================================
extern "C" void kernel_launch(void* const* d_in, const int* in_sizes, int n_in,
                              void* d_out, int out_size, void* d_ws, size_t ws_size,
                              hipStream_t stream) {
  const float* x      = (const float*)d_in[0];   // (B,S,D) f32
  const float* source = (const float*)d_in[1];   // (B,S,N) f32
  const int*   posids = (const int*)d_in[2];     // (B,S) int
  // d_in[3]=r, d_in[4]=window_size (compile-time constants here)
  const float* W      = (const float*)d_in[5];   // (D,G) f32

  // ---- workspace layout (~6.4 MB) ----
  float* gn    = (float*)d_ws;                              // 16384*64 f32
  float* norms = gn + (size_t)(Bn * Sn) * Gn;               // 16384 f32
  unsigned long long* keys =
      (unsigned long long*)(norms + (size_t)Bn * Sn);       // B*PPAD u64
  int* partner = (int*)(keys + (size_t)Bn * PPAD);          // B*S i32
  int* keepmap = partner + Bn * Sn;                         // B*KEEP i32

  // ---- output layout: x_m | s_m | kept position ids ----
  float* out_x = (float*)d_out;
  float* out_s = out_x + (size_t)Bn * KEEP * Dn;
  int*   out_p = (int*)(out_s + (size_t)Bn * KEEP * Nn);

  // 1) WMMA GEMM + normalize: 16384 rows / (8 waves * 16 rows) = 128 blocks;
  //    dynamic LDS holds W with padded stride.
  gemm_norm_kernel<<<(Bn * Sn) / 128, 256,
                     Dn * WLDSTRIDE * sizeof(float), stream>>>(x, W, gn, norms);

  // 2) pair similarities -> sort keys
  sim_kernel<<<(Bn * PPAD) / 256, 256, 0, stream>>>(gn, keys);

  // 3) bitonic sort, LDS-fused: 15 launches instead of 136 global passes
  bitonic_local_sort_kernel<<<(Bn * PPAD) / TB, 1024, 0, stream>>>(keys);
  for (int k = TB * 2; k <= PPAD; k <<= 1) {
    for (int j = k >> 1; j >= TB; j >>= 1)
      bitonic_global_kernel<<<(Bn * PPAD) / 256, 256, 0, stream>>>(keys, k, j);
    bitonic_local_merge_kernel<<<(Bn * PPAD) / TB, 1024, 0, stream>>>(keys, k);
  }

  // 4) sequential greedy matching + keep map
  greedy_kernel<<<Bn, 256, 0, stream>>>(keys, partner, keepmap);

  // 5) merge x + gather position ids
  merge_x_kernel<<<Bn * KEEP, 128, 0, stream>>>(x, norms, partner, keepmap,
                                                posids, out_x, out_p);

  // 6) merge source (bandwidth-dominant streaming pass)
  merge_s_kernel<<<Bn * KEEP, 256, 0, stream>>>(source, partner, keepmap, out_s);
}